// HetroGAT_28587302322738
// MI455X (gfx1250) — compile-verified
//
#include <hip/hip_runtime.h>
#include <math.h>

// Problem constants (from reference)
#define NN    20000
#define EE    160000
#define RR    3
#define LL    2
#define IN_   64
#define HIDD  128
#define HH    16
#define FHH   8
#define HD2_  64

typedef __attribute__((ext_vector_type(2))) float v2f;
typedef __attribute__((ext_vector_type(8))) float v8f;

// ---------------- utility: zero fill ----------------
__global__ void fill0_kernel(float* __restrict__ p, int n) {
  int i = blockIdx.x * blockDim.x + threadIdx.x;
  for (; i < n; i += gridDim.x * blockDim.x) p[i] = 0.f;
}

// ---------------- WMMA fp32 GEMM, LDS-staged, fragment-packed B ----------------
// C[MxNc] = A[MxK] @ B[KxNc] (+bias) (+skip)
// Block = 128 threads (4 waves) -> 64 rows x 32 cols of C.
// Wave w computes rows [base+16w, +16), two 16x16 col tiles (dual accumulators).
// A fragment (16x4 f32, ISA 7.12.2): lane lo holds row M=lo;
//   VGPR0 = K=2*hi, VGPR1 = K=2*hi+1   (hi = lane>>4)  -> contiguous in row-major A
// B stored in LDS pre-packed so each fragment {B[k][c], B[k+1][c]} is one
// aligned 8-byte ds_load_b64 straight into an even VGPR pair (no re-pack movs):
//   sB[(kp*32 + c)*2 + j] = B[2*kp + j][c0 + c]
// C/D (16x16 f32): VGPR v -> M = v + 8*hi, N = lo
template <int K>
__global__ void gemm_wmma_lds_kernel(const float* __restrict__ A,
                                     const float* __restrict__ B,
                                     const float* __restrict__ bias,
                                     const float* __restrict__ skip,
                                     float* __restrict__ C,
                                     int M, int Nc) {
  constexpr int LDA = K + 4;  // pad: fragment reads spread across LDS banks;
                              // also keeps row byte-stride a multiple of 16
  __shared__ float sA[64 * LDA];
  __shared__ float sB[K * 32];  // packed fragment-major (see above)

  const int tid  = threadIdx.x;
  const int wave = tid >> 5;
  const int lane = tid & 31;
  const int lo = lane & 15;
  const int hi = lane >> 4;
  const int rowBase = blockIdx.y * 64;
  const int c0 = blockIdx.x * 32;

  // stage A panel: float4 global loads -> float4 LDS stores (coalesced)
  for (int idx = tid; idx < 64 * (K / 4); idx += 128) {
    int r  = idx / (K / 4);
    int c4 = idx - r * (K / 4);
    int gr = rowBase + r;
    if (gr >= M) gr = 0;  // clamp OOB rows; results masked at store
    float4 val = *(const float4*)(A + (size_t)gr * K + c4 * 4);
    *(float4*)&sA[r * LDA + c4 * 4] = val;
  }
  // stage B panel: read 2 adjacent K-rows as float2, pack-interleave, one float4 store
  for (int idx = tid; idx < K * 8; idx += 128) {
    int kp = idx >> 4;         // K-row pair index
    int c  = (idx & 15) * 2;   // column pair within 32-wide tile
    const float* b0p = B + (size_t)(2 * kp) * Nc + (c0 + c);
    float2 r0 = *(const float2*)b0p;
    float2 r1 = *(const float2*)(b0p + Nc);
    float4 w;
    w.x = r0.x; w.y = r1.x;    // fragment for col c
    w.z = r0.y; w.w = r1.y;    // fragment for col c+1
    *(float4*)&sB[(kp * 32 + c) * 2] = w;
  }
  __syncthreads();

  v8f acc0 = {0.f, 0.f, 0.f, 0.f, 0.f, 0.f, 0.f, 0.f};
  v8f acc1 = {0.f, 0.f, 0.f, 0.f, 0.f, 0.f, 0.f, 0.f};
  const float* __restrict__ saw = sA + (wave * 16 + lo) * LDA;

#if __has_builtin(__builtin_amdgcn_wmma_f32_16x16x4_f32)
#pragma unroll
  for (int k = 0; k < K; k += 4) {
    // A fragment: 8B aligned (LDA even, k+2*hi even) -> single ds_load_b64
    v2f a = *(const v2f*)&saw[k + 2 * hi];
    // B fragments: packed -> single aligned ds_load_b64 each
    const float* bb = sB + (k / 2 + hi) * 64;  // (kp)*32*2
    v2f b0 = *(const v2f*)&bb[lo * 2];
    v2f b1 = *(const v2f*)&bb[32 + lo * 2];
    // (neg_a, A, neg_b, B, c_mod, C, reuse_a, reuse_b)
    acc0 = __builtin_amdgcn_wmma_f32_16x16x4_f32(false, a, false, b0, (short)0,
                                                 acc0, false, false);
    acc1 = __builtin_amdgcn_wmma_f32_16x16x4_f32(false, a, false, b1, (short)0,
                                                 acc1, false, false);
  }
#else
  for (int v = 0; v < 8; ++v) {
    const int r = wave * 16 + v + 8 * hi;
    float s0 = 0.f, s1 = 0.f;
    for (int k = 0; k < K; ++k) {
      float bv0 = sB[((k >> 1) * 32 + lo) * 2 + (k & 1)];
      float bv1 = sB[((k >> 1) * 32 + 16 + lo) * 2 + (k & 1)];
      s0 += sA[r * LDA + k] * bv0;
      s1 += sA[r * LDA + k] * bv1;
    }
    acc0[v] = s0;
    acc1[v] = s1;
  }
#endif

  const int r0 = rowBase + wave * 16;
#pragma unroll
  for (int v = 0; v < 8; ++v) {
    const int row = r0 + v + 8 * hi;
    if (row < M) {
      const int col0 = c0 + lo;
      const int col1 = c0 + 16 + lo;
      float x0 = acc0[v];
      float x1 = acc1[v];
      if (bias) { x0 += bias[col0]; x1 += bias[col1]; }
      const size_t i0 = (size_t)row * Nc + col0;
      const size_t i1 = (size_t)row * Nc + col1;
      if (skip) { x0 += skip[i0]; x1 += skip[i1]; }  // same-thread RMW: safe if skip==C
      C[i0] = x0;
      C[i1] = x1;
    }
  }
}

// ---------------- BatchNorm: per-column sum / sumsq ----------------
__global__ void colstats_kernel(const float* __restrict__ X, int M, int C,
                                float* __restrict__ sums, int rowsPerChunk) {
  const int col = threadIdx.x;  // blockDim.x == C
  const int row0 = blockIdx.x * rowsPerChunk;
  int rend = row0 + rowsPerChunk;
  if (rend > M) rend = M;
  float s = 0.f, sq = 0.f;
  for (int r = row0; r < rend; ++r) {
    float x = X[(size_t)r * C + col];  // coalesced across threads
    s += x;
    sq += x * x;
  }
  atomicAdd(&sums[col], s);
  atomicAdd(&sums[C + col], sq);
}

__global__ void bn_finalize_kernel(const float* __restrict__ sums, int M, int C,
                                   const float* __restrict__ g,
                                   const float* __restrict__ bt,
                                   float* __restrict__ scale_shift) {
  int c = threadIdx.x;
  if (c >= C) return;
  float inv = 1.f / (float)M;
  float mu = sums[c] * inv;
  float var = sums[C + c] * inv - mu * mu;  // biased variance (training mode)
  float rs = rsqrtf(var + 1e-5f);
  float sc = g[c] * rs;
  scale_shift[c] = sc;
  scale_shift[C + c] = bt[c] - sc * mu;
}

__global__ void bn_relu_kernel(const float* __restrict__ X,
                               const float* __restrict__ ss,
                               float* __restrict__ Y, int total, int C) {
  int i = blockIdx.x * blockDim.x + threadIdx.x;
  for (; i < total; i += gridDim.x * blockDim.x) {
    int c = i % C;
    float v = X[i] * ss[c] + ss[C + c];
    Y[i] = v > 0.f ? v : 0.f;
  }
}

// ---------------- GAT: per-node attention logits ----------------
__global__ void elr_kernel(const float* __restrict__ z,
                           const float* __restrict__ al,
                           const float* __restrict__ ar,
                           float* __restrict__ el, float* __restrict__ er) {
  int i = blockIdx.x * blockDim.x + threadIdx.x;  // N*H
  if (i >= NN * HH) return;
  int n = i >> 4, h = i & 15;
  const float* zp = z + (size_t)n * HIDD + h * FHH;
  const float* alp = al + h * FHH;
  const float* arp = ar + h * FHH;
  float sl = 0.f, sr = 0.f;
#pragma unroll
  for (int f = 0; f < FHH; ++f) {
    float zv = zp[f];
    sl += zv * alp[f];
    sr += zv * arp[f];
  }
  el[i] = sl;
  er[i] = sr;
}

// float -> order-preserving uint mapping (for atomic segment max)
__device__ __forceinline__ unsigned fmap(float f) {
  unsigned u = __float_as_uint(f);
  return (u >> 31) ? ~u : (u | 0x80000000u);
}
__device__ __forceinline__ float funmap(unsigned u) {
  unsigned b = (u >> 31) ? (u & 0x7FFFFFFFu) : ~u;
  return __uint_as_float(b);
}

__global__ void edge_max_kernel(const float* __restrict__ el,
                                const float* __restrict__ er,
                                const int* __restrict__ src,
                                const int* __restrict__ dst,
                                float* __restrict__ mmax,  // holds mapped uints
                                float* __restrict__ ebuf) {
  int i = blockIdx.x * blockDim.x + threadIdx.x;
  if (i >= EE * HH) return;
  int e = i >> 4, h = i & 15;
  int s = src[e], d = dst[e];
  float v = el[s * HH + h] + er[d * HH + h];
  v = v > 0.f ? v : 0.2f * v;  // leaky_relu 0.2
  ebuf[i] = v;
  atomicMax(reinterpret_cast<unsigned*>(mmax) + d * HH + h, fmap(v));
}

__global__ void mfix_kernel(float* __restrict__ m) {
  int i = blockIdx.x * blockDim.x + threadIdx.x;
  if (i >= NN * HH) return;
  float f = funmap(__float_as_uint(m[i]));
  m[i] = isfinite(f) ? f : 0.f;  // empty segments -> 0 (matches reference)
}

__global__ void edge_exp_kernel(float* __restrict__ ebuf,
                                const int* __restrict__ dst,
                                const float* __restrict__ m,
                                float* __restrict__ s) {
  int i = blockIdx.x * blockDim.x + threadIdx.x;
  if (i >= EE * HH) return;
  int e = i >> 4, h = i & 15;
  int d = dst[e];
  float ex = expf(ebuf[i] - m[d * HH + h]);
  ebuf[i] = ex;  // in place: e -> exp(e - m[dst])
  atomicAdd(&s[d * HH + h], ex);
}

__global__ void edge_agg_kernel(const float* __restrict__ ebuf,
                                const float* __restrict__ s,
                                const float* __restrict__ z,
                                const int* __restrict__ src,
                                const int* __restrict__ dst,
                                float* __restrict__ agg) {
  int i = blockIdx.x * blockDim.x + threadIdx.x;
  if (i >= EE * HH) return;
  int e = i >> 4, h = i & 15;
  int sn = src[e], d = dst[e];
  float alpha = ebuf[i] / s[d * HH + h];
  const float* zp = z + (size_t)sn * HIDD + h * FHH;
  float* ap = agg + (size_t)d * HIDD + h * FHH;
#pragma unroll
  for (int f = 0; f < FHH; ++f) atomicAdd(&ap[f], alpha * zp[f]);
}

// v += (agg + gat_b) * softmax(attn_p, axis=relations)[r]
__global__ void relcombine_kernel(const float* __restrict__ agg,
                                  const float* __restrict__ gb,
                                  const float* __restrict__ attn_p_l, int r,
                                  float* __restrict__ v) {
  int i = blockIdx.x * blockDim.x + threadIdx.x;
  const int total = NN * HIDD;
  for (; i < total; i += gridDim.x * blockDim.x) {
    int f = i % HIDD;
    float p0 = attn_p_l[f];
    float p1 = attn_p_l[HIDD + f];
    float p2 = attn_p_l[2 * HIDD + f];
    float mx = fmaxf(p0, fmaxf(p1, p2));
    float e0 = expf(p0 - mx), e1 = expf(p1 - mx), e2 = expf(p2 - mx);
    float w = (r == 0 ? e0 : (r == 1 ? e1 : e2)) / (e0 + e1 + e2);
    v[i] += (agg[i] + gb[f]) * w;
  }
}

// final 64 -> 1 projection
__global__ void decode_out_kernel(const float* __restrict__ Hact,
                                  const float* __restrict__ W,
                                  const float* __restrict__ b,
                                  float* __restrict__ out) {
  int n = blockIdx.x * blockDim.x + threadIdx.x;
  if (n >= NN) return;
  const float* hp = Hact + (size_t)n * HD2_;
  float sum = 0.f;
#pragma unroll
  for (int j = 0; j < HD2_; ++j) sum += hp[j] * W[j];
  out[n] = sum + b[0];
}

// ---------------- orchestration ----------------
extern "C" void kernel_launch(void* const* d_in, const int* in_sizes, int n_in,
                              void* d_out, int out_size, void* d_ws, size_t ws_size,
                              hipStream_t stream) {
  (void)in_sizes; (void)n_in; (void)out_size; (void)ws_size;
  const float* inputs = (const float*)d_in[0];
  const int*   src    = (const int*)d_in[1];
  const int*   dst    = (const int*)d_in[2];
  const float* e1_W = (const float*)d_in[3];
  const float* e1_b = (const float*)d_in[4];
  const float* e_g  = (const float*)d_in[5];
  const float* e_bt = (const float*)d_in[6];
  const float* e2_W = (const float*)d_in[7];
  const float* e2_b = (const float*)d_in[8];
  const float* gat_W  = (const float*)d_in[9];
  const float* gat_al = (const float*)d_in[10];
  const float* gat_ar = (const float*)d_in[11];
  const float* gat_b  = (const float*)d_in[12];
  const float* attn_p = (const float*)d_in[13];
  const float* m1_W = (const float*)d_in[14];
  const float* m1_b = (const float*)d_in[15];
  const float* m_g  = (const float*)d_in[16];
  const float* m_bt = (const float*)d_in[17];
  const float* m2_W = (const float*)d_in[18];
  const float* m2_b = (const float*)d_in[19];
  const float* d1_W = (const float*)d_in[20];
  const float* d1_b = (const float*)d_in[21];
  const float* d_g  = (const float*)d_in[22];
  const float* d_bt = (const float*)d_in[23];
  const float* d2_W = (const float*)d_in[24];
  const float* d2_b = (const float*)d_in[25];
  float* out = (float*)d_out;

  // workspace layout (floats): ~77 MB total; all GEMM operands 16B-aligned
  float* Wf = (float*)d_ws;
  const size_t NH  = (size_t)NN * HIDD;  // 2.56M
  const size_t NHh = (size_t)NN * HH;    // 0.32M
  float* h    = Wf;
  float* z    = Wf + NH;
  float* agg  = Wf + 2 * NH;
  float* v    = Wf + 3 * NH;
  float* t1   = Wf + 4 * NH;
  float* t2   = Wf + 5 * NH;
  float* el   = Wf + 6 * NH;
  float* er   = el + NHh;
  float* mb   = er + NHh;
  float* sb   = mb + NHh;
  float* exb  = sb + NHh;                // E*H
  float* stats = exb + (size_t)EE * HH;  // 4*128 floats

  auto gemm = [&](const float* A, const float* B, const float* bias,
                  const float* skip, float* C, int K, int Ncols) {
    dim3 grid(Ncols / 32, (NN + 63) / 64);
    if (K == 128)
      gemm_wmma_lds_kernel<128><<<grid, 128, 0, stream>>>(A, B, bias, skip, C, NN, Ncols);
    else
      gemm_wmma_lds_kernel<64><<<grid, 128, 0, stream>>>(A, B, bias, skip, C, NN, Ncols);
  };
  auto zero = [&](float* p, size_t n) {
    int blocks = (int)((n + 255) / 256);
    if (blocks > 2048) blocks = 2048;
    fill0_kernel<<<blocks, 256, 0, stream>>>(p, (int)n);
  };
  auto bnstage = [&](float* X, int C, const float* g, const float* bt, float* Y) {
    zero(stats, 2 * (size_t)C);
    const int rpc = 160;
    colstats_kernel<<<(NN + rpc - 1) / rpc, C, 0, stream>>>(X, NN, C, stats, rpc);
    bn_finalize_kernel<<<1, C, 0, stream>>>(stats, NN, C, g, bt, stats + 2 * C);
    int total = NN * C;
    int blocks = (total + 255) / 256;
    if (blocks > 4096) blocks = 4096;
    bn_relu_kernel<<<blocks, 256, 0, stream>>>(X, stats + 2 * C, Y, total, C);
  };

  const int EH = EE * HH;
  const int NHhI = NN * HH;

  // ---- Embed MLP (skip=True: adds hidden) ----
  gemm(inputs, e1_W, e1_b, nullptr, t1, IN_, HIDD);
  bnstage(t1, HIDD, e_g, e_bt, t2);
  gemm(t2, e2_W, e2_b, t2, h, HIDD, HIDD);

  // ---- L layers of relational GAT + MLP residual ----
  for (int l = 0; l < LL; ++l) {
    zero(v, NH);
    for (int r = 0; r < RR; ++r) {
      const float* Wlr = gat_W + (size_t)(l * RR + r) * HIDD * HIDD;
      gemm(h, Wlr, nullptr, nullptr, z, HIDD, HIDD);
      elr_kernel<<<(NHhI + 255) / 256, 256, 0, stream>>>(
          z, gat_al + (size_t)(l * RR + r) * HH * FHH,
          gat_ar + (size_t)(l * RR + r) * HH * FHH, el, er);
      zero(mb, NHh);
      zero(sb, NHh);
      zero(agg, NH);
      const int* sr = src + (size_t)r * EE;
      const int* ds = dst + (size_t)r * EE;
      edge_max_kernel<<<(EH + 255) / 256, 256, 0, stream>>>(el, er, sr, ds, mb, exb);
      mfix_kernel<<<(NHhI + 255) / 256, 256, 0, stream>>>(mb);
      edge_exp_kernel<<<(EH + 255) / 256, 256, 0, stream>>>(exb, ds, mb, sb);
      edge_agg_kernel<<<(EH + 255) / 256, 256, 0, stream>>>(exb, sb, z, sr, ds, agg);
      relcombine_kernel<<<4096, 256, 0, stream>>>(
          agg, gat_b + (size_t)(l * RR + r) * HIDD,
          attn_p + (size_t)l * RR * HIDD, r, v);
    }
    // layer MLP (no internal skip) + residual h
    gemm(v, m1_W + (size_t)l * HIDD * HD2_, m1_b + l * HD2_, nullptr, t1, HIDD, HD2_);
    bnstage(t1, HD2_, m_g + l * HD2_, m_bt + l * HD2_, t2);
    gemm(t2, m2_W + (size_t)l * HD2_ * HIDD, m2_b + l * HIDD, h, h, HD2_, HIDD);
  }

  // ---- Decode MLP ----
  gemm(h, d1_W, d1_b, nullptr, t1, HIDD, HD2_);
  bnstage(t1, HD2_, d_g, d_bt, t2);
  decode_out_kernel<<<(NN + 255) / 256, 256, 0, stream>>>(t2, d2_W, d2_b, out);
}